// GATLayer_59030030516650
// MI455X (gfx1250) — compile-verified
//
#include <hip/hip_runtime.h>
#include <math.h>

#define DIM 128
#define MIN_NORM 1e-10f
#define BALL_EPS 1e-5f
#define CRW 0.1f

typedef __attribute__((ext_vector_type(2))) float v2f;
typedef __attribute__((ext_vector_type(8))) float v8f;

__device__ __forceinline__ float wave_sum(float v) {
#pragma unroll
  for (int off = 16; off > 0; off >>= 1) v += __shfl_xor(v, off, 32);
  return v;
}
__device__ __forceinline__ float dot4(float4 a, float4 b) {
  return a.x * b.x + a.y * b.y + a.z * b.z + a.w * b.w;
}
__device__ __forceinline__ float4 scale4(float4 a, float s) {
  return make_float4(a.x * s, a.y * s, a.z * s, a.w * s);
}
// monotone float<->uint order-preserving transform for atomicMax-based segment max
__device__ __forceinline__ unsigned flip_f32(float f) {
  unsigned u = __float_as_uint(f);
  return (u & 0x80000000u) ? ~u : (u | 0x80000000u);
}
__device__ __forceinline__ float unflip_f32(unsigned u) {
  unsigned b = (u & 0x80000000u) ? (u & 0x7FFFFFFFu) : ~u;
  return __uint_as_float(b);
}

// --- fused log_map_zero + [16 x 128] @ [128 x 128] WMMA f32 GEMM -------------
// One workgroup = 16 output rows; 8 waves each own a 16-col slab (whole N=128).
__global__ __launch_bounds__(256) void tangent_gemm_kernel(
    const float* __restrict__ X, const float* __restrict__ W,
    float* __restrict__ OUT, float* __restrict__ TAN, int M) {
  __shared__ float As[16][DIM + 4];
  const int row0 = blockIdx.x * 16;
  const int t = threadIdx.x;
  // stage 16x128 rows into LDS (zero-pad past M)
  for (int i = t; i < 16 * DIM; i += 256) {
    int r = i / DIM, c = i % DIM, gr = row0 + r;
    As[r][c] = (gr < M) ? X[(size_t)gr * DIM + c] : 0.0f;
  }
  __syncthreads();
  const int wave = t >> 5, lane = t & 31;
  // log_map_zero in place: 8 waves x 2 rows, wave32 shuffle reduction per row
#pragma unroll
  for (int rr = 0; rr < 2; ++rr) {
    int r = wave * 2 + rr;
    float4 v = *(const float4*)&As[r][lane * 4];
    float ss = wave_sum(dot4(v, v));
    float n = fminf(fmaxf(sqrtf(ss), MIN_NORM), 1.0f - BALL_EPS);
    float s = atanhf(n) / n;
    *(float4*)&As[r][lane * 4] = scale4(v, s);
  }
  __syncthreads();
  if (TAN) {  // rels path also needs tangent vectors
    for (int i = t; i < 16 * DIM; i += 256) {
      int r = i / DIM, c = i % DIM, gr = row0 + r;
      if (gr < M) TAN[(size_t)gr * DIM + c] = As[r][c];
    }
  }
  // WMMA f32 16x16x4, K = 128 in 32 steps.
  // A 16x4 layout: lanes 0-15 -> M=lane, K={k0,k0+1}; lanes 16-31 -> K={k0+2,k0+3}
  // B 4x16 layout mirrors A (rows K striped, cols N across lanes)
  const int n0 = wave * 16;
  const int m = lane & 15;
  const int kb = (lane >> 4) * 2;
  v8f c = {};
#pragma unroll
  for (int k0 = 0; k0 < DIM; k0 += 4) {
    v2f a, b;
    a.x = As[m][k0 + kb];
    a.y = As[m][k0 + kb + 1];
    b.x = W[(size_t)(k0 + kb) * DIM + n0 + m];
    b.y = W[(size_t)(k0 + kb + 1) * DIM + n0 + m];
    c = __builtin_amdgcn_wmma_f32_16x16x4_f32(false, a, false, b, (short)0, c,
                                              false, false);
  }
  // C/D layout: VGPR j -> row j (lanes 0-15) / row j+8 (lanes 16-31), col = lane&15
  const int mbase = (lane >> 4) * 8;
#pragma unroll
  for (int j = 0; j < 8; ++j) {
    int gr = row0 + mbase + j;
    if (gr < M) OUT[(size_t)gr * DIM + n0 + m] = c[j];
  }
}

// --- edge pass A: scores + segment max (flip-transformed u32 atomicMax) -----
__global__ __launch_bounds__(256) void edge_scores_kernel(
    const float* __restrict__ E, const int* __restrict__ src,
    const int* __restrict__ dst, float* __restrict__ scores,
    unsigned* __restrict__ row_max, int nE) {
  int e = (blockIdx.x * blockDim.x + threadIdx.x) >> 5;
  int lane = threadIdx.x & 31;
  if (e >= nE) return;
  int s = src[e], d = dst[e];
  float4 a = ((const float4*)(E + (size_t)s * DIM))[lane];
  float4 b = ((const float4*)(E + (size_t)d * DIM))[lane];
  float p = wave_sum(dot4(a, b));
  if (lane == 0) {
    scores[e] = p;
    atomicMax(&row_max[s], flip_f32(p));
  }
}

// --- edge pass B: exp(score - rowmax), segment sum ---------------------------
__global__ __launch_bounds__(256) void edge_exp_kernel(
    const int* __restrict__ src, float* __restrict__ scores,
    const unsigned* __restrict__ row_max, float* __restrict__ row_sum, int nE) {
  int e = blockIdx.x * blockDim.x + threadIdx.x;
  if (e >= nE) return;
  float m = unflip_f32(row_max[src[e]]);   // sentinel 0 -> NaN -> 0 (matches ref)
  if (!isfinite(m)) m = 0.0f;
  float ex = expf(scores[e] - m);
  scores[e] = ex;
  atomicAdd(&row_sum[src[e]], ex);
}

// --- edge pass C: acc1[src] += alpha * E[dst] (wave per edge) ---------------
__global__ __launch_bounds__(256) void edge_scatter_kernel(
    const float* __restrict__ E, const int* __restrict__ src,
    const int* __restrict__ dst, const float* __restrict__ ex,
    const float* __restrict__ row_sum, float* __restrict__ acc, int nE) {
  int e = (blockIdx.x * blockDim.x + threadIdx.x) >> 5;
  int lane = threadIdx.x & 31;
  if (e >= nE) return;
  int s = src[e], d = dst[e];
  float alpha = ex[e] / fmaxf(row_sum[s], MIN_NORM);
  float4 v = ((const float4*)(E + (size_t)d * DIM))[lane];
  float* o = acc + (size_t)s * DIM + lane * 4;
  atomicAdd(o + 0, alpha * v.x);
  atomicAdd(o + 1, alpha * v.y);
  atomicAdd(o + 2, alpha * v.z);
  atomicAdd(o + 3, alpha * v.w);
}

// --- acc2[rel_ent] += Rm[rel_id] (wave per incidence; Rm is L2-hot 512KB) ---
__global__ __launch_bounds__(256) void rel_scatter_kernel(
    const float* __restrict__ Rm, const int* __restrict__ rent,
    const int* __restrict__ rid, float* __restrict__ acc, int nE) {
  int e = (blockIdx.x * blockDim.x + threadIdx.x) >> 5;
  int lane = threadIdx.x & 31;
  if (e >= nE) return;
  int en = rent[e], r = rid[e];
  float4 v = ((const float4*)(Rm + (size_t)r * DIM))[lane];
  float* o = acc + (size_t)en * DIM + lane * 4;
  atomicAdd(o + 0, v.x);
  atomicAdd(o + 1, v.y);
  atomicAdd(o + 2, v.z);
  atomicAdd(o + 3, v.w);
}

// --- rel_id histogram via LDS (replaces a 1.6M x 128 scatter) ---------------
__global__ __launch_bounds__(256) void count_rels_kernel(
    const int* __restrict__ rid, int nE, unsigned* __restrict__ counts, int R) {
  __shared__ unsigned hist[1024];
  for (int i = threadIdx.x; i < 1024; i += 256) hist[i] = 0u;
  __syncthreads();
  for (int i = blockIdx.x * blockDim.x + threadIdx.x; i < nE;
       i += gridDim.x * blockDim.x)
    atomicAdd(&hist[rid[i] & 1023], 1u);
  __syncthreads();
  for (int i = threadIdx.x; i < R; i += 256)
    if (hist[i]) atomicAdd(&counts[i], hist[i]);
}

// --- hyperbolic helpers (wave-per-row; all 32 lanes participate) ------------
__device__ __forceinline__ float4 proj4(float4 v) {
  float n = fmaxf(sqrtf(wave_sum(dot4(v, v))), MIN_NORM);
  return scale4(v, fminf(1.0f, (1.0f - BALL_EPS) / n));
}
__device__ __forceinline__ float4 expmap_proj4(float4 v) {
  float n = fmaxf(sqrtf(wave_sum(dot4(v, v))), MIN_NORM);
  float s = tanhf(n) / n;                       // exp_map_zero scale
  float nn = fmaxf(n * s, MIN_NORM);            // norm after exp-map
  float f = fminf(1.0f, (1.0f - BALL_EPS) / nn);
  return scale4(v, s * f);
}
__device__ __forceinline__ float4 mobius_proj4(float4 x, float4 y) {
  float xy = wave_sum(dot4(x, y));
  float x2 = wave_sum(dot4(x, x));
  float y2 = wave_sum(dot4(y, y));
  float ca = 1.0f + 2.0f * xy + y2;
  float cb = 1.0f - x2;
  float den = fmaxf(1.0f + 2.0f * xy + x2 * y2, MIN_NORM);
  float4 r = make_float4((ca * x.x + cb * y.x) / den, (ca * x.y + cb * y.y) / den,
                         (ca * x.z + cb * y.z) / den, (ca * x.w + cb * y.w) / den);
  return proj4(r);
}

__global__ __launch_bounds__(256) void finalize_ents_kernel(
    const float* __restrict__ acc1, const float* __restrict__ acc2,
    const float* __restrict__ deg, const float* __restrict__ bias,
    float* __restrict__ out, int N) {
  int row = blockIdx.x * 8 + (threadIdx.x >> 5);
  int lane = threadIdx.x & 31;
  if (row >= N) return;
  float4 u = ((const float4*)(acc1 + (size_t)row * DIM))[lane];
  float4 x = expmap_proj4(u);                                  // ents_near_ents
  float inv = 1.0f / deg[row];
  float4 w = scale4(((const float4*)(acc2 + (size_t)row * DIM))[lane], inv);
  float4 y = scale4(expmap_proj4(w), CRW);                     // 0.1 * ents_near_rels
  float4 z = mobius_proj4(x, y);
  float4 bh = expmap_proj4(((const float4*)bias)[lane]);
  float4 o = mobius_proj4(z, bh);
  ((float4*)(out + (size_t)row * DIM))[lane] = o;
}

__global__ __launch_bounds__(256) void finalize_rels_kernel(
    const float* __restrict__ relsT, const unsigned* __restrict__ counts,
    const float* __restrict__ num, float* __restrict__ out, int R) {
  int row = blockIdx.x * 8 + (threadIdx.x >> 5);
  int lane = threadIdx.x & 31;
  if (row >= R) return;
  float ratio = (float)counts[row] / num[row];   // segsum(rels_t[rid]) = count*rels_t
  float4 t = scale4(((const float4*)(relsT + (size_t)row * DIM))[lane], ratio);
  ((float4*)(out + (size_t)row * DIM))[lane] = expmap_proj4(t);
}

extern "C" void kernel_launch(void* const* d_in, const int* in_sizes, int n_in,
                              void* d_out, int out_size, void* d_ws, size_t ws_size,
                              hipStream_t stream) {
  const float* ents_embed = (const float*)d_in[0];
  const float* rels_embed = (const float*)d_in[1];
  const float* W_ent = (const float*)d_in[2];
  const float* W_rel = (const float*)d_in[3];
  const float* bias = (const float*)d_in[4];
  const float* ent_deg = (const float*)d_in[5];   // ents_near_rels_num
  const float* rel_num = (const float*)d_in[6];   // rels_near_ents_num
  const int* ent_src = (const int*)d_in[7];
  const int* ent_dst = (const int*)d_in[8];
  const int* rel_ent = (const int*)d_in[9];
  const int* rel_id = (const int*)d_in[10];
  const int N = in_sizes[0] / DIM;
  const int R = in_sizes[1] / DIM;
  const int nEE = in_sizes[7];
  const int nER = in_sizes[9];

  // workspace layout
  float* ws = (float*)d_ws;
  float* E = ws;            ws += (size_t)N * DIM;
  float* acc1 = ws;         ws += (size_t)N * DIM;   // ents_near_ents tangent acc
  float* acc2 = ws;         ws += (size_t)N * DIM;   // ents_near_rels tangent acc
  float* Rm = ws;           ws += (size_t)R * DIM;
  float* relsT = ws;        ws += (size_t)R * DIM;
  float* scores = ws;       ws += (size_t)nEE;       // scores, then exp in place
  unsigned* row_max = (unsigned*)ws;  ws += N;       // flip-encoded, 0 = sentinel
  float* row_sum = (float*)ws;        ws += N;
  unsigned* counts = (unsigned*)ws;   ws += R;

  hipMemsetAsync(acc1, 0, (size_t)2 * N * DIM * sizeof(float), stream);
  hipMemsetAsync(row_max, 0, (size_t)(2 * N + R) * sizeof(unsigned), stream);

  tangent_gemm_kernel<<<(N + 15) / 16, 256, 0, stream>>>(ents_embed, W_ent, E,
                                                         nullptr, N);
  tangent_gemm_kernel<<<(R + 15) / 16, 256, 0, stream>>>(rels_embed, W_rel, Rm,
                                                         relsT, R);
  edge_scores_kernel<<<(nEE + 7) / 8, 256, 0, stream>>>(E, ent_src, ent_dst,
                                                        scores, row_max, nEE);
  edge_exp_kernel<<<(nEE + 255) / 256, 256, 0, stream>>>(ent_src, scores,
                                                         row_max, row_sum, nEE);
  edge_scatter_kernel<<<(nEE + 7) / 8, 256, 0, stream>>>(
      E, ent_src, ent_dst, scores, row_sum, acc1, nEE);
  rel_scatter_kernel<<<(nER + 7) / 8, 256, 0, stream>>>(Rm, rel_ent, rel_id,
                                                        acc2, nER);
  count_rels_kernel<<<512, 256, 0, stream>>>(rel_id, nER, counts, R);

  finalize_ents_kernel<<<(N + 7) / 8, 256, 0, stream>>>(
      acc1, acc2, ent_deg, bias, (float*)d_out, N);
  finalize_rels_kernel<<<(R + 7) / 8, 256, 0, stream>>>(
      relsT, counts, rel_num, (float*)d_out + (size_t)N * DIM, R);
}